// GraphAttentionHead_26895085207862
// MI455X (gfx1250) — compile-verified
//
#include <hip/hip_runtime.h>
#include <math.h>

typedef float v2f __attribute__((ext_vector_type(2)));
typedef float v8f __attribute__((ext_vector_type(8)));

#define NEG_SLOPE 0.2f
#define NB 8
#define NN 2048
#define DD 256

// ---------------------------------------------------------------------------
// Kernel 1: Wh = x @ W^T using V_WMMA_F32_16X16X4_F32.
// Also writes Wh_t[b][col][n] (transposed) for WMMA-B-friendly b64 loads in
// the attention kernel; each lane's 8 D-values are consecutive in n, so the
// transposed store coalesces into b128s per lane.
// Block = 512 threads = 16 waves; tile = 16 rows x 256 cols.
// ---------------------------------------------------------------------------
__global__ __launch_bounds__(512)
void wh_kernel(const float* __restrict__ x, const float* __restrict__ W,
               float* __restrict__ Wh, float* __restrict__ Wh_t) {
    __shared__ float xs[16 * 260];   // stride 260 -> conflict-free b64 reads
    const int tid  = threadIdx.x;
    const int wave = tid >> 5;
    const int lane = tid & 31;
    const int lm   = lane & 15;
    const int hi   = lane >> 4;
    const int row0 = blockIdx.x * 16;
    const int b    = row0 >> 11;          // NN = 2048 rows per batch
    const int n0   = row0 & (NN - 1);

    for (int idx = tid; idx < 16 * 256; idx += 512) {
        int r = idx >> 8, c = idx & 255;
        xs[r * 260 + c] = x[(size_t)(row0 + r) * DD + c];
    }
    __syncthreads();

    const int col0 = wave * 16;
    const float* wrow = W + (size_t)(col0 + lm) * DD;
    v8f acc = {};
    for (int k = 0; k < DD; k += 4) {
        v2f a, bf;
        a.x  = xs[lm * 260 + k + 2 * hi];
        a.y  = xs[lm * 260 + k + 2 * hi + 1];
        bf.x = wrow[k + 2 * hi];
        bf.y = wrow[k + 2 * hi + 1];
        acc = __builtin_amdgcn_wmma_f32_16x16x4_f32(false, a, false, bf,
                                                    (short)0, acc, false, false);
    }
    // Row-major store (coalesced over lm)
    float* outp = Wh + (size_t)row0 * DD + col0 + lm;
    // Transposed store: 8 consecutive floats per lane -> merged wide stores
    float* outt = Wh_t + ((size_t)b * DD + col0 + lm) * NN + n0 + 8 * hi;
#pragma unroll
    for (int v = 0; v < 8; ++v) {
        outp[(size_t)(v + 8 * hi) * DD] = acc[v];
        outt[v] = acc[v];
    }
}

// ---------------------------------------------------------------------------
// Kernel 2: s_i = Wh . a1, s_j = Wh . a2 (one wave32 per row, shuffle reduce)
// ---------------------------------------------------------------------------
__global__ __launch_bounds__(256)
void score_kernel(const float* __restrict__ Wh, const float* __restrict__ a,
                  float* __restrict__ s_i, float* __restrict__ s_j) {
    const int wave = threadIdx.x >> 5;
    const int lane = threadIdx.x & 31;
    const int row  = blockIdx.x * 8 + wave;
    const float* p = Wh + (size_t)row * DD;
    float p1 = 0.f, p2 = 0.f;
    for (int c = lane; c < DD; c += 32) {
        float w = p[c];
        p1 += w * a[c];
        p2 += w * a[DD + c];
    }
#pragma unroll
    for (int off = 16; off > 0; off >>= 1) {
        p1 += __shfl_xor(p1, off, 32);
        p2 += __shfl_xor(p2, off, 32);
    }
    if (lane == 0) { s_i[row] = p1; s_j[row] = p2; }
}

// ---------------------------------------------------------------------------
// Kernel 3: masked softmax + alpha_nb @ Wh + fused epilogue.
// Block = 256 threads = 8 waves; tile = 32 rows of one batch (two 16-row
// alpha tiles sharing B fragments). Wave w owns cols [32w, 32w+32) ->
// four v8f accumulators; inner k-step = 2 ds_b64 + 2 global_b64 + 4 WMMA.
// ---------------------------------------------------------------------------
__global__ __launch_bounds__(256)
void attn_kernel(const float* __restrict__ adj, const float* __restrict__ Wh,
                 const float* __restrict__ Wh_t,
                 const float* __restrict__ s_i, const float* __restrict__ s_j,
                 float* __restrict__ out) {
    __shared__ float sj_s[NN];
    __shared__ float red[256];
    __shared__ float m_s[32], sum_s[32], asel_s[32], si_s[32];
    __shared__ float alpha_s[32 * 68];   // stride 68 -> conflict-free A reads

    const int tid  = threadIdx.x;
    const int wave = tid >> 5;
    const int lane = tid & 31;
    const int lm   = lane & 15;
    const int hi   = lane >> 4;
    const int bt   = blockIdx.x;
    const int b    = bt >> 6;            // 64 row-tiles (of 32) per batch
    const int i0   = (bt & 63) * 32;

    const float* adjb = adj  + (size_t)b * NN * NN;
    const float* Whb  = Wh   + (size_t)b * NN * DD;
    const float* Wtb  = Wh_t + (size_t)b * DD * NN;

    for (int j = tid; j < NN; j += 256) sj_s[j] = s_j[(size_t)b * NN + j];
    if (tid < 32) si_s[tid] = s_i[(size_t)b * NN + i0 + tid];
    __syncthreads();

    const int r  = tid >> 3;             // row in tile 0..31
    const int c  = tid & 7;              // 8 scanners per row
    const int gi = i0 + r;
    const float si = si_s[r];
    const float* arow = adjb + (size_t)gi * NN;

    // ---- pass 1a: row max over included edges (self-loop forced)
    float m = -INFINITY;
    for (int j = c; j < NN; j += 8) {
        float av = arow[j];
        if (av != 0.f || j == gi) m = fmaxf(m, si + sj_s[j]);
    }
    red[tid] = m;
    __syncthreads();
    if (c == 0) {
        float mm = -INFINITY;
#pragma unroll
        for (int q = 0; q < 8; ++q) mm = fmaxf(mm, red[r * 8 + q]);
        m_s[r] = mm;
    }
    __syncthreads();
    const float mrow = m_s[r];

    // ---- pass 1b: row sum of exp
    float s = 0.f;
    for (int j = c; j < NN; j += 8) {
        float av = arow[j];
        if (av != 0.f || j == gi) s += __expf(si + sj_s[j] - mrow);
    }
    red[tid] = s;
    __syncthreads();
    if (c == 0) {
        float ss = 0.f;
#pragma unroll
        for (int q = 0; q < 8; ++q) ss += red[r * 8 + q];
        sum_s[r]  = ss;
        asel_s[r] = __expf(si_s[r] + sj_s[i0 + r] - m_s[r]) / ss;  // alpha_self
    }
    __syncthreads();

    // ---- pass 2: acc = alpha_nb @ Wh via f32 WMMA, K chunks of 64
    const int col0 = wave * 32;
    v8f acc00 = {}, acc01 = {}, acc10 = {}, acc11 = {};
    for (int jc = 0; jc < NN; jc += 64) {
        // Prefetch next chunk of adj (32 rows x 256B = 64 cachelines)
        if (jc + 64 < NN && tid < 64) {
            __builtin_prefetch(&adjb[(size_t)(i0 + (tid >> 1)) * NN +
                                     jc + 64 + (tid & 1) * 32], 0, 1);
        }
        // Build alpha chunk (diagonal zeroed), coalesced over jj.
        for (int idx = tid; idx < 32 * 64; idx += 256) {
            int rr = idx >> 6, jj = idx & 63;
            int j = jc + jj;
            int girr = i0 + rr;
            float av  = adjb[(size_t)girr * NN + j];
            float val = 0.f;
            if (av != 0.f && j != girr)
                val = __expf(si_s[rr] + sj_s[j] - m_s[rr]) / sum_s[rr];
            alpha_s[rr * 68 + jj] = val;
        }
        __syncthreads();
        const float* pb0 = Wtb + (size_t)(col0 + lm) * NN + jc + 2 * hi;
        const float* pb1 = Wtb + (size_t)(col0 + 16 + lm) * NN + jc + 2 * hi;
#pragma unroll 4
        for (int kk = 0; kk < 64; kk += 4) {
            v2f a0, a1, b0, b1;
            a0.x = alpha_s[lm * 68 + kk + 2 * hi];
            a0.y = alpha_s[lm * 68 + kk + 2 * hi + 1];
            a1.x = alpha_s[(16 + lm) * 68 + kk + 2 * hi];
            a1.y = alpha_s[(16 + lm) * 68 + kk + 2 * hi + 1];
            // B fragment: B[k][n] = Wh_t[col0+n][jc+k] -> single b64 per lane
            b0.x = pb0[kk];  b0.y = pb0[kk + 1];
            b1.x = pb1[kk];  b1.y = pb1[kk + 1];
            acc00 = __builtin_amdgcn_wmma_f32_16x16x4_f32(false, a0, false, b0,
                                                          (short)0, acc00, false, false);
            acc01 = __builtin_amdgcn_wmma_f32_16x16x4_f32(false, a0, false, b1,
                                                          (short)0, acc01, false, false);
            acc10 = __builtin_amdgcn_wmma_f32_16x16x4_f32(false, a1, false, b0,
                                                          (short)0, acc10, false, false);
            acc11 = __builtin_amdgcn_wmma_f32_16x16x4_f32(false, a1, false, b1,
                                                          (short)0, acc11, false, false);
        }
        __syncthreads();
    }

    // ---- epilogue: out = alpha_self * Wh + leaky_relu(neighbor_sum)
#pragma unroll
    for (int v = 0; v < 8; ++v) {
        int rr0 = v + 8 * hi;        // rows 0..15 tile
        int rr1 = 16 + rr0;          // rows 16..31 tile
        float as0 = asel_s[rr0];
        float as1 = asel_s[rr1];
        const float* wp0 = Whb + (size_t)(i0 + rr0) * DD + col0 + lm;
        const float* wp1 = Whb + (size_t)(i0 + rr1) * DD + col0 + lm;
        size_t o0 = ((size_t)b * NN + i0 + rr0) * DD + col0 + lm;
        size_t o1 = ((size_t)b * NN + i0 + rr1) * DD + col0 + lm;
        float n00 = acc00[v], n01 = acc01[v], n10 = acc10[v], n11 = acc11[v];
        float c00 = (n00 >= 0.f) ? n00 : NEG_SLOPE * n00;
        float c01 = (n01 >= 0.f) ? n01 : NEG_SLOPE * n01;
        float c10 = (n10 >= 0.f) ? n10 : NEG_SLOPE * n10;
        float c11 = (n11 >= 0.f) ? n11 : NEG_SLOPE * n11;
        out[o0]      = as0 * wp0[0]  + c00;
        out[o0 + 16] = as0 * wp0[16] + c01;
        out[o1]      = as1 * wp1[0]  + c10;
        out[o1 + 16] = as1 * wp1[16] + c11;
    }
}

// ---------------------------------------------------------------------------
extern "C" void kernel_launch(void* const* d_in, const int* in_sizes, int n_in,
                              void* d_out, int out_size, void* d_ws, size_t ws_size,
                              hipStream_t stream) {
    const float* x   = (const float*)d_in[0];   // (8, 2048, 256)
    const float* adj = (const float*)d_in[1];   // (8, 2048, 2048)
    const float* W   = (const float*)d_in[2];   // (256, 256)
    const float* a   = (const float*)d_in[3];   // (1, 512)
    float* out = (float*)d_out;                 // (8, 2048, 256)

    float* Wh   = (float*)d_ws;                 // 8*2048*256 floats = 16 MB
    float* Wh_t = Wh   + (size_t)NB * NN * DD;  // 16 MB (transposed per batch)
    float* s_i  = Wh_t + (size_t)NB * NN * DD;  // 16384 floats
    float* s_j  = s_i  + (size_t)NB * NN;       // 16384 floats

    wh_kernel   <<<NB * NN / 16, 512, 0, stream>>>(x, W, Wh, Wh_t);
    score_kernel<<<NB * NN / 8,  256, 0, stream>>>(Wh, a, s_i, s_j);
    attn_kernel <<<NB * NN / 32, 256, 0, stream>>>(adj, Wh, Wh_t, s_i, s_j, out);
}